// SelfAttentiveSpanExtractor_38087769981168
// MI455X (gfx1250) — compile-verified
//
#include <hip/hip_runtime.h>
#include <hip/hip_bf16.h>
#include <math.h>

// Problem constants (from reference): B=8, S=4096, N=1024, D=1024,
// NUM_WIDTH=64, WIDTH_DIM=64, MAX_WIDTH=32 -> span length <= 32.
#define B_ 8
#define S_ 4096
#define N_ 1024
#define D_ 1024
#define NUM_WIDTH_ 64
#define WIDTH_DIM_ 64
#define OUTD_ (D_ + WIDTH_DIM_) // 1088

typedef float v2f __attribute__((ext_vector_type(2)));
typedef float v8f __attribute__((ext_vector_type(8)));

// ---------------------------------------------------------------------------
// Pass 1: logits[b,s] = dot(seq[b,s,:], w) + bias.  One wave32 per token.
// Fully coalesced float4 loads, butterfly wave reduction.
// ---------------------------------------------------------------------------
__global__ __launch_bounds__(256) void sase_logits_kernel(
    const float* __restrict__ seq, const float* __restrict__ w,
    const float* __restrict__ bias, float* __restrict__ logits) {
  const int tid  = threadIdx.x;
  const int lane = tid & 31;
  const long t   = (long)blockIdx.x * 8 + (tid >> 5); // token in [0, B*S)

  const float4* row = (const float4*)(seq + t * (long)D_);
  const float4* wv  = (const float4*)w;

  float sum = 0.f;
#pragma unroll
  for (int i = 0; i < 8; ++i) {           // D/4 = 256 float4 / 32 lanes = 8
    float4 a = row[lane + 32 * i];
    float4 b = wv[lane + 32 * i];
    sum += a.x * b.x + a.y * b.y + a.z * b.z + a.w * b.w;
  }
#pragma unroll
  for (int m = 16; m >= 1; m >>= 1) sum += __shfl_xor(sum, m, 32);
  if (lane == 0) logits[t] = sum + bias[0];
}

// ---------------------------------------------------------------------------
// Pass 2: one block (8 waves) per span.
//   wave 0 : masked softmax over <=32 logits -> sw[] in LDS
//   wave k : 128-wide D slice; weighted row-sum via V_WMMA_F32_16X16X4_F32.
//     A[m,k] = w[k]  (broadcast across all 16 rows; documented A layout:
//                     K = vgpr + 2*(lane>>4))
//     B[k,n] = seq[b, start+k, d0+n]  (N = lane&15 across lanes,
//                                      K = vgpr + 2*(lane>>4))
//   => every row of D equals the weighted sum; lane l (l<16) holds the
//      result for d0+l in accumulator VGPR 0 (constant index, no LDS
//      bounce), giving one coalesced 64B store per tile.
// ---------------------------------------------------------------------------
__global__ __launch_bounds__(256) void sase_span_kernel(
    const float* __restrict__ seq, const int* __restrict__ spans,
    const float* __restrict__ logits, const float* __restrict__ wemb,
    float* __restrict__ out) {
  const int bn   = blockIdx.x;      // [0, B*N)
  const int b    = bn >> 10;        // / N_
  const int tid  = threadIdx.x;
  const int lane = tid & 31;
  const int wave = tid >> 5;

  const int start = spans[2 * bn + 0];
  const int end   = spans[2 * bn + 1];
  const bool valid = (start >= 0) && (end >= start);
  int widthc = end - start;
  if (widthc < 0) widthc = 0;
  if (widthc > NUM_WIDTH_ - 1) widthc = NUM_WIDTH_ - 1;

  __shared__ float sw[32];

  if (tid < 32) {  // wave 0: masked softmax over the span (length <= 32)
    const int k = lane;
    const bool inSpan = valid && (start + k <= end);
    float lg = -INFINITY;
    if (inSpan) lg = logits[(long)b * S_ + start + k];
    float m = lg;
#pragma unroll
    for (int x = 16; x >= 1; x >>= 1) m = fmaxf(m, __shfl_xor(m, x, 32));
    float e = inSpan ? expf(lg - m) : 0.f;
    float s = e;
#pragma unroll
    for (int x = 16; x >= 1; x >>= 1) s += __shfl_xor(s, x, 32);
    sw[k] = (valid && s > 0.f) ? (e / s) : 0.f;
  }
  __syncthreads();

  const int half = lane >> 4;       // 0: lanes 0-15, 1: lanes 16-31
  const int m16  = lane & 15;       // N index within tile (data matrix)
  const long outbase = (long)bn * OUTD_;
  const float* seqb = seq + (long)b * (long)S_ * D_;

  const int baseRow = valid ? start : 0;
  const int lastRow = valid ? end : 0;   // clamp keeps loads in-bounds;
                                         // out-of-span k has weight 0.
  const int dwave = wave * 128;

#pragma unroll
  for (int tile = 0; tile < 8; ++tile) {
    const int d0 = dwave + tile * 16;
    v8f c = {0.f, 0.f, 0.f, 0.f, 0.f, 0.f, 0.f, 0.f};
#pragma unroll
    for (int k0 = 0; k0 < 32; k0 += 4) {
      const int kx = k0 + 2 * half;           // K for VGPR 0 of A/B
      int rx = baseRow + kx;     if (rx > lastRow) rx = lastRow;
      int ry = baseRow + kx + 1; if (ry > lastRow) ry = lastRow;
      v2f aw, bd;
      aw.x = sw[kx];                          // A: weights, K = vgpr + 2*half
      aw.y = sw[kx + 1];                      //    (broadcast over M rows)
      bd.x = seqb[(long)rx * D_ + d0 + m16];  // B: data, N = lane&15
      bd.y = seqb[(long)ry * D_ + d0 + m16];  //    K = vgpr + 2*half
      c = __builtin_amdgcn_wmma_f32_16x16x4_f32(
          /*neg_a=*/false, aw, /*neg_b=*/false, bd,
          /*c_mod=*/(short)0, c, /*reuse_a=*/false, /*reuse_b=*/false);
    }
    // D rows all identical; lane l<16 holds out[d0+l] in accumulator VGPR 0.
    if (half == 0) {
      out[outbase + d0 + m16] = c[0];
    }
  }

  // Width embedding (written regardless of `valid`, as in the reference).
  if (tid < WIDTH_DIM_) {
    out[outbase + D_ + tid] = wemb[widthc * WIDTH_DIM_ + tid];
  }
}

// ---------------------------------------------------------------------------
extern "C" void kernel_launch(void* const* d_in, const int* in_sizes, int n_in,
                              void* d_out, int out_size, void* d_ws,
                              size_t ws_size, hipStream_t stream) {
  (void)in_sizes; (void)n_in; (void)out_size; (void)ws_size;
  const float* seq   = (const float*)d_in[0];  // (B,S,D) f32
  const int*   spans = (const int*)d_in[1];    // (B,N,2) int32
  const float* attw  = (const float*)d_in[2];  // (1,D) f32
  const float* attb  = (const float*)d_in[3];  // (1,)  f32
  const float* wemb  = (const float*)d_in[4];  // (64,64) f32
  float* out    = (float*)d_out;               // (B,N,1088) f32
  float* logits = (float*)d_ws;                // B*S floats = 128 KB scratch

  sase_logits_kernel<<<(B_ * S_) / 8, 256, 0, stream>>>(seq, attw, attb, logits);
  sase_span_kernel<<<B_ * N_, 256, 0, stream>>>(seq, spans, logits, wemb, out);
}